// InvarPDEs_Net_RS_77884936945784
// MI455X (gfx1250) — compile-verified
//
#include <hip/hip_runtime.h>
#include <hip/hip_bf16.h>
#include <math.h>

// ---------------------------------------------------------------------------
// Types for CDNA5 WMMA (wave32). f32 16x16x4: A,B = 2 VGPRs (v2f), C/D = 8 (v8f)
// ---------------------------------------------------------------------------
typedef float v2f __attribute__((ext_vector_type(2)));
typedef float v8f __attribute__((ext_vector_type(8)));

#define BLK 256

// Gaussian-kernel constants (EPS_RBF = 0.99)
#define E2   0.9801f
#define T2E2 1.9602f
#define F4E2 3.84238404f
#define BN_EPS 1e-5f

// ---------------------------------------------------------------------------
// u init: u[c][p] = x[b][c%3][hw],  p = b*4096 + hw, N0 = 64*64*64
// ---------------------------------------------------------------------------
__global__ void init_u_kernel(const float* __restrict__ x, float* __restrict__ u) {
  const int N0 = 64 * 64 * 64;
  int idx = blockIdx.x * BLK + threadIdx.x;   // over 25*N0
  int c  = idx / N0;
  int p  = idx - c * N0;
  int b  = p >> 12;
  int hw = p & 4095;
  u[idx] = x[(b * 3 + (c % 3)) * 4096 + hw];
}

__global__ void zero_fill_kernel(float* __restrict__ p, int n) {
  int idx = blockIdx.x * BLK + threadIdx.x;
  if (idx < n) p[idx] = 0.f;
}

// Copy W[M][K] -> Wp[Mpad][Kpad] with zero padding (removes all predication
// from the GEMM inner loop).
__global__ void pad_weights_kernel(const float* __restrict__ W, float* __restrict__ Wp,
                                   int M, int K, int Kpad, int total) {
  int idx = blockIdx.x * BLK + threadIdx.x;   // over Mpad*Kpad
  if (idx >= total) return;
  int m = idx / Kpad;
  int k = idx - m * Kpad;
  Wp[idx] = (m < M && k < K) ? W[m * K + k] : 0.f;
}

// ---------------------------------------------------------------------------
// 7x7 depthwise derivative convs (zero padded, cross-correlation like XLA).
// kdx[i][j] = 2*e2*(i-3)*g   kdy[i][j] = -2*e2*(j-3)*g
// kdxx[i][j] = (4*e2^2*(i-3)^2-2*e2)*g   kdyy: (j-3)^2
// ---------------------------------------------------------------------------
__global__ void deriv4_kernel(const float* __restrict__ u,
                              float* __restrict__ ux, float* __restrict__ uy,
                              float* __restrict__ uxx, float* __restrict__ uyy,
                              int Nn, int H, int W, int HW) {
  int idx = blockIdx.x * BLK + threadIdx.x;   // over C*Nn (exact)
  int p  = idx % Nn;
  int hw = p % HW;
  int y  = hw / W;
  int x  = hw - y * W;
  int rowbase = idx - hw;                     // c*Nn + b*HW
  float ex[7];
#pragma unroll
  for (int d = 0; d < 7; ++d) { float dd = (float)(d - 3); ex[d] = expf(-E2 * dd * dd); }
  float sx = 0.f, sy = 0.f, sxx = 0.f, syy = 0.f;
#pragma unroll
  for (int i = 0; i < 7; ++i) {
    int dy = i - 3, yy = y + dy;
    if ((unsigned)yy < (unsigned)H) {
#pragma unroll
      for (int j = 0; j < 7; ++j) {
        int dx = j - 3, xx = x + dx;
        if ((unsigned)xx < (unsigned)W) {
          float v = u[rowbase + yy * W + xx];
          float g = ex[i] * ex[j];
          float fdy = (float)dy, fdx = (float)dx;
          sx  += v * (T2E2 * fdy) * g;
          sy  += v * (-T2E2 * fdx) * g;
          sxx += v * (F4E2 * fdy * fdy - T2E2) * g;
          syy += v * (F4E2 * fdx * fdx - T2E2) * g;
        }
      }
    }
  }
  ux[idx] = sx; uy[idx] = sy; uxx[idx] = sxx; uyy[idx] = syy;
}

// uxy = conv(uy, kdx)  (second padded pass — matches reference boundaries)
__global__ void derivxy_kernel(const float* __restrict__ uy, float* __restrict__ uxy,
                               int Nn, int H, int W, int HW) {
  int idx = blockIdx.x * BLK + threadIdx.x;
  int p  = idx % Nn;
  int hw = p % HW;
  int y  = hw / W;
  int x  = hw - y * W;
  int rowbase = idx - hw;
  float ex[7];
#pragma unroll
  for (int d = 0; d < 7; ++d) { float dd = (float)(d - 3); ex[d] = expf(-E2 * dd * dd); }
  float s = 0.f;
#pragma unroll
  for (int i = 0; i < 7; ++i) {
    int dy = i - 3, yy = y + dy;
    if ((unsigned)yy < (unsigned)H) {
#pragma unroll
      for (int j = 0; j < 7; ++j) {
        int dx = j - 3, xx = x + dx;
        if ((unsigned)xx < (unsigned)W) {
          float v = uy[rowbase + yy * W + xx];
          s += v * (T2E2 * (float)dy) * (ex[i] * ex[j]);
        }
      }
    }
  }
  uxy[idx] = s;
}

// ---------------------------------------------------------------------------
// Invariants.  inv rows: [0,C)=inv1(u)  [C,2C)=inv3  [2C,3C)=inv4  [3C,4C)=inv5
//              [4C,5C)=inv2  [5C,6C-1)=inv6.  Also per-sample group max-abs.
// ---------------------------------------------------------------------------
__global__ void zero_scales_kernel(unsigned* __restrict__ smax) {
  if (threadIdx.x < 192) smax[threadIdx.x] = 0u;
}

__global__ void inv_kernel(const float* __restrict__ u,  const float* __restrict__ ux,
                           const float* __restrict__ uy, const float* __restrict__ uxx,
                           const float* __restrict__ uyy, const float* __restrict__ uxy,
                           float* __restrict__ inv, unsigned* __restrict__ smax,
                           int C, int Nn, int hwShift) {
  int idx = blockIdx.x * BLK + threadIdx.x;   // over C*Nn (exact, multiple of 256)
  int c = idx / Nn;
  int p = idx - c * Nn;
  int b = p >> hwShift;

  float uv  = u[idx];
  float vx  = ux[idx],  vy  = uy[idx];
  float vxx = uxx[idx], vyy = uyy[idx], vxy = uxy[idx];
  float xx = vx * vx, yy = vy * vy, xy = vx * vy;
  float xyxy = xy * vxy;

  float i3 = xx * vxx + 2.f * xyxy + yy * vyy;
  float i4 = yy * vxx - 2.f * xyxy + xx * vyy;
  float i5 = xy * (vyy - vxx) + (xx - yy) * vxy;
  float i2 = xx + yy;

  inv[(0 * C + c) * Nn + p] = uv;
  inv[(1 * C + c) * Nn + p] = i3;
  inv[(2 * C + c) * Nn + p] = i4;
  inv[(3 * C + c) * Nn + p] = i5;
  inv[(4 * C + c) * Nn + p] = i2;

  float m0 = fabsf(uv);
  float m1 = fmaxf(fabsf(i3), fmaxf(fabsf(i4), fabsf(i5)));
  float m2 = fabsf(i2);
  if (c < C - 1) {
    float i6 = ux[idx + Nn] * vx + uy[idx + Nn] * vy;
    inv[(5 * C + c) * Nn + p] = i6;
    m2 = fmaxf(m2, fabsf(i6));
  }
  // wave32 max-reduce (all lanes of a wave share b: HW >= 64)
#pragma unroll
  for (int off = 16; off > 0; off >>= 1) {
    m0 = fmaxf(m0, __shfl_xor(m0, off, 32));
    m1 = fmaxf(m1, __shfl_xor(m1, off, 32));
    m2 = fmaxf(m2, __shfl_xor(m2, off, 32));
  }
  if ((threadIdx.x & 31) == 0) {
    atomicMax(&smax[0 * 64 + b], __float_as_uint(m0));
    atomicMax(&smax[1 * 64 + b], __float_as_uint(m1));
    atomicMax(&smax[2 * 64 + b], __float_as_uint(m2));
  }
}

__global__ void recip_scales_kernel(const unsigned* __restrict__ smax,
                                    float* __restrict__ srec) {
  if (threadIdx.x < 192) srec[threadIdx.x] = 1.0f / __uint_as_float(smax[threadIdx.x]);
}

// ---------------------------------------------------------------------------
// WMMA fp32 GEMM:  Y[M][N] = Wp[Mpad][Kpad] * Bop(X)[Kpad][N]
//   MODE 0: Bop = identity
//   MODE 1: Bop = X * scale[group(k)][sample]  (sample is wave-uniform per chunk)
// One wave -> 16 (M) x 64 (N); K stepped by 4 via V_WMMA_F32_16X16X4_F32.
// All inner-loop loads are unpredicated (operands zero-padded in memory);
// row pointers are strength-reduced (no per-step 64-bit multiplies).
// A/B lane layout: row/col = lane%16, K = vgpr + 2*(lane>=16).
// ---------------------------------------------------------------------------
template <int MODE>
__global__ void gemm16_kernel(const float* __restrict__ Wp, const float* __restrict__ X,
                              float* __restrict__ Y, int M, int Kpad, int Nn,
                              const float* __restrict__ srec, int Cg, int hwShift) {
  int wave   = (blockIdx.x * BLK + threadIdx.x) >> 5;
  int mtiles = (M + 15) >> 4;
  int total  = mtiles * (Nn >> 6);
  if (wave >= total) return;                    // wave-uniform exit
  int mt   = wave % mtiles;
  int n0   = (wave / mtiles) << 6;
  int lane = threadIdx.x & 31;
  int half = lane >> 4;
  int lm   = lane & 15;
  int rowA = (mt << 4) + lm;                    // always < Mpad (padded)

  float s0 = 0.f, s1 = 0.f, s2 = 0.f;
  if (MODE == 1) {
    int bs = __builtin_amdgcn_readfirstlane(n0 >> hwShift);  // sample, uniform
    s0 = srec[bs]; s1 = srec[64 + bs]; s2 = srec[128 + bs];
  }

  const v8f z = {0.f, 0.f, 0.f, 0.f, 0.f, 0.f, 0.f, 0.f};
  v8f acc0 = z, acc1 = z, acc2 = z, acc3 = z;

  // strength-reduced pointers
  const float* Ap  = Wp + (size_t)rowA * Kpad + (half << 1);  // A row, k = k0
  const float* Xp  = X + (size_t)(half << 1) * Nn + n0;       // B row k0
  const size_t rowStep = 4 * (size_t)Nn;
  int k0 = half << 1;

  int kSteps = Kpad >> 2;
  for (int ks = 0; ks < kSteps; ++ks) {
    v2f a = *(const v2f*)Ap;                    // global_load_b64, unpredicated
    const float* XpN = Xp + Nn;                 // B row k0+1

    // all 8 B loads up front so they clause together
    v2f b0, b1, b2, b3;
    b0.x = Xp[lm];       b0.y = XpN[lm];
    b1.x = Xp[16 + lm];  b1.y = XpN[16 + lm];
    b2.x = Xp[32 + lm];  b2.y = XpN[32 + lm];
    b3.x = Xp[48 + lm];  b3.y = XpN[48 + lm];

    if (MODE == 1) {
      float sc0 = (k0 < Cg) ? s0 : ((k0 < 4 * Cg) ? s1 : s2);
      int k1 = k0 + 1;
      float sc1 = (k1 < Cg) ? s0 : ((k1 < 4 * Cg) ? s1 : s2);
      b0.x *= sc0; b0.y *= sc1;
      b1.x *= sc0; b1.y *= sc1;
      b2.x *= sc0; b2.y *= sc1;
      b3.x *= sc0; b3.y *= sc1;
    }

    // cheap clamped prefetch of next k-step's B rows (no u64 multiply)
    size_t poff = (ks + 1 < kSteps) ? rowStep : (size_t)0;
    __builtin_prefetch(Xp + poff + lane * 2, 0, 0);

    acc0 = __builtin_amdgcn_wmma_f32_16x16x4_f32(false, a, false, b0, (short)0, acc0, false, false);
    acc1 = __builtin_amdgcn_wmma_f32_16x16x4_f32(false, a, false, b1, (short)0, acc1, false, false);
    acc2 = __builtin_amdgcn_wmma_f32_16x16x4_f32(false, a, false, b2, (short)0, acc2, false, false);
    acc3 = __builtin_amdgcn_wmma_f32_16x16x4_f32(false, a, false, b3, (short)0, acc3, false, false);

    Ap += 4;
    Xp += rowStep;
    if (MODE == 1) k0 += 4;
  }

  // D layout: M = vgpr + 8*(lane>=16), N = lane%16
  int mbase = (mt << 4) + (half << 3);
#pragma unroll
  for (int r = 0; r < 8; ++r) {
    int m = mbase + r;
    if (m < M) {
      float* Yr = Y + (size_t)m * Nn + n0 + lm;
      Yr[0]  = acc0[r];
      Yr[16] = acc1[r];
      Yr[32] = acc2[r];
      Yr[48] = acc3[r];
    }
  }
}

// ---------------------------------------------------------------------------
// BatchNorm statistics over N per channel (deterministic fixed-order 2 stage).
// ---------------------------------------------------------------------------
__global__ void bn_stats_kernel(const float* __restrict__ X, float* __restrict__ pS,
                                float* __restrict__ pQ, int M, int Nn) {
  int m = blockIdx.y;
  int chunk = blockIdx.x;                 // 64 chunks
  int len = Nn >> 6;
  int start = chunk * len;
  float s = 0.f, q = 0.f;
  for (int i = threadIdx.x; i < len; i += BLK) {
    float v = X[(size_t)m * Nn + start + i];
    s += v; q += v * v;
  }
  __shared__ float ss[BLK], sq[BLK];
  ss[threadIdx.x] = s; sq[threadIdx.x] = q;
  __syncthreads();
  for (int o = BLK / 2; o > 0; o >>= 1) {
    if (threadIdx.x < o) { ss[threadIdx.x] += ss[threadIdx.x + o]; sq[threadIdx.x] += sq[threadIdx.x + o]; }
    __syncthreads();
  }
  if (threadIdx.x == 0) { pS[m * 64 + chunk] = ss[0]; pQ[m * 64 + chunk] = sq[0]; }
}

__global__ void bn_finalize_kernel(const float* __restrict__ pS, const float* __restrict__ pQ,
                                   const float* __restrict__ gamma, const float* __restrict__ beta,
                                   float* __restrict__ aOut, float* __restrict__ cOut,
                                   int M, int Nn) {
  int m = threadIdx.x;
  if (m >= M) return;
  float S = 0.f, Q = 0.f;
  for (int i = 0; i < 64; ++i) { S += pS[m * 64 + i]; Q += pQ[m * 64 + i]; }
  float inv_n = 1.0f / (float)Nn;
  float mean  = S * inv_n;
  float var   = Q * inv_n - mean * mean;
  float rs    = rsqrtf(var + BN_EPS);
  float g     = gamma[m] * rs;
  aOut[m] = g;
  cOut[m] = beta[m] - mean * g;
}

// In-place BN+ReLU transform: X = relu(a[ch]*X + c[ch])  (over M*Nn elements)
__global__ void bnrelu_inplace_kernel(float* __restrict__ X, const float* __restrict__ a,
                                      const float* __restrict__ c, int Nn, int total) {
  int idx = blockIdx.x * BLK + threadIdx.x;
  if (idx >= total) return;
  int ch = idx / Nn;
  X[idx] = fmaxf(a[ch] * X[idx] + c[ch], 0.f);
}

// Lift tail: BN + ReLU + 2x2 maxpool (monotone => order-free)
__global__ void pool_bn_relu_kernel(const float* __restrict__ pre, float* __restrict__ u,
                                    const float* __restrict__ a, const float* __restrict__ c,
                                    int Nold, int Nnew, int HWold, int HWnew,
                                    int Wold, int Wnew) {
  int idx = blockIdx.x * BLK + threadIdx.x;   // over M2*Nnew
  int m  = idx / Nnew;
  int pn = idx - m * Nnew;
  int b  = pn / HWnew;
  int hw = pn - b * HWnew;
  int yn = hw / Wnew;
  int xn = hw - yn * Wnew;
  int base = m * Nold + b * HWold;
  float am = a[m], cm = c[m];
  float v = -1e30f;
#pragma unroll
  for (int dy = 0; dy < 2; ++dy)
#pragma unroll
    for (int dx = 0; dx < 2; ++dx) {
      float t = pre[base + (2 * yn + dy) * Wold + (2 * xn + dx)];
      v = fmaxf(v, fmaxf(am * t + cm, 0.f));
    }
  u[idx] = v;
}

// u = relu(u + step[t] * (a2*h2 + c2))
__global__ void update_u_kernel(float* __restrict__ u, const float* __restrict__ h2,
                                const float* __restrict__ a, const float* __restrict__ c,
                                const float* __restrict__ stepArr, int t, int Nn) {
  int idx = blockIdx.x * BLK + threadIdx.x;
  int ch = idx / Nn;
  float v = u[idx] + stepArr[t] * (a[ch] * h2[idx] + c[ch]);
  u[idx] = fmaxf(v, 0.f);
}

// feat[b][c] = max over 64 spatial of u (C=200, HW=64)
__global__ void feat_kernel(const float* __restrict__ u, float* __restrict__ feat) {
  int idx = blockIdx.x * BLK + threadIdx.x;   // 64*200
  if (idx >= 64 * 200) return;
  int b = idx / 200;
  int c = idx - b * 200;
  const float* p = u + c * 4096 + b * 64;
  float m = -1e30f;
#pragma unroll 8
  for (int i = 0; i < 64; ++i) m = fmaxf(m, p[i]);
  feat[idx] = m;
}

// fc1 -> relu -> fc2 -> log_softmax ; one block (64 thr) per sample
__global__ void head_kernel(const float* __restrict__ feat,
                            const float* __restrict__ w1, const float* __restrict__ b1,
                            const float* __restrict__ w2, const float* __restrict__ b2,
                            float* __restrict__ out) {
  int b = blockIdx.x;
  int j = threadIdx.x;   // 64
  __shared__ float ly[64];
  __shared__ float lz[10];
  __shared__ float logZ;
  float s = 0.f;
  for (int k = 0; k < 200; ++k) s += w1[j * 200 + k] * feat[b * 200 + k];
  ly[j] = fmaxf(s + b1[j], 0.f);
  __syncthreads();
  if (j < 10) {
    float s2 = 0.f;
    for (int k = 0; k < 64; ++k) s2 += w2[j * 64 + k] * ly[k];
    lz[j] = s2 + b2[j];
  }
  __syncthreads();
  if (j == 0) {
    float mx = -1e30f;
    for (int i = 0; i < 10; ++i) mx = fmaxf(mx, lz[i]);
    float se = 0.f;
    for (int i = 0; i < 10; ++i) se += expf(lz[i] - mx);
    logZ = mx + logf(se);
  }
  __syncthreads();
  if (j < 10) out[b * 10 + j] = lz[j] - logZ;
}

// ---------------------------------------------------------------------------
// Host orchestration
// ---------------------------------------------------------------------------
static inline int gemm_blocks(int M, int Nn) {
  int waves = ((M + 15) >> 4) * (Nn >> 6);
  return (waves + 7) / 8;   // 8 waves / 256-thread block
}
static inline int ceil_div(int a, int b) { return (a + b - 1) / b; }
static inline int pad4(int k) { return (k + 3) & ~3; }
static inline int pad16(int m) { return (m + 15) & ~15; }

extern "C" void kernel_launch(void* const* d_in, const int* in_sizes, int n_in,
                              void* d_out, int out_size, void* d_ws, size_t ws_size,
                              hipStream_t stream) {
  (void)out_size; (void)ws_size;
  // ---- parameter leaf indexing (robust to flatten-order ambiguity) ----
  int xIdx, pb;
  if (in_sizes[0] == 64 * 3 * 64 * 64) { xIdx = 0; pb = 1; }
  else                                 { xIdx = n_in - 1; pb = 0; }
  bool sortedKeys = (in_sizes[pb + 128] == 64 * 200);  // fc1.w right after blocks?
  int i_fc1w, i_fc1b, i_fc2w, i_fc2b, i_lift0, i_step;
  if (sortedKeys) { i_fc1w = pb + 128; i_fc1b = pb + 129; i_fc2w = pb + 130;
                    i_fc2b = pb + 131; i_lift0 = pb + 132; i_step = pb + 144; }
  else            { i_lift0 = pb + 128; i_fc1w = pb + 140; i_fc1b = pb + 141;
                    i_fc2w = pb + 142; i_fc2b = pb + 143; i_step = pb + 144; }
  auto FP = [&](int i) -> const float* { return (const float*)d_in[i]; };

  // ---- workspace carve (floats) ----
  float* ws = (float*)d_ws;
  const size_t OFF_U    = 0;                   // <= 28*262144 = 7.35M (padded)
  const size_t OFF_DV   = 7400000;             // 5 deriv fields; h1/h2 alias later
  const size_t OFF_H2   = OFF_DV + 14000000;
  const size_t OFF_INV  = 40400000;            // <= 152*262144 = 39.85M (padded)
  const size_t OFF_PART = 80300000;            // pS/pQ 2*200*64
  const size_t OFF_WP   = 80326000;            // padded weights <= 208*1200
  const size_t OFF_BNA  = 80576000, OFF_BNC  = 80576256;
  const size_t OFF_BNA2 = 80576512, OFF_BNC2 = 80576768;
  const size_t OFF_SMAX = 80577024;            // 192 uints
  const size_t OFF_SREC = 80577216;            // 192
  const size_t OFF_FEAT = 80577408;            // 12800

  float*    u    = ws + OFF_U;
  float*    h1   = ws + OFF_DV;        // aliases derivs (dead by GEMM time)
  float*    h2   = ws + OFF_H2;
  float*    inv  = ws + OFF_INV;
  float*    pS   = ws + OFF_PART;
  float*    pQ   = ws + OFF_PART + 200 * 64;
  float*    Wp   = ws + OFF_WP;
  float*    bnA  = ws + OFF_BNA;  float* bnC  = ws + OFF_BNC;
  float*    bnA2 = ws + OFF_BNA2; float* bnC2 = ws + OFF_BNC2;
  unsigned* smax = (unsigned*)(ws + OFF_SMAX);
  float*    srec = ws + OFF_SREC;
  float*    feat = ws + OFF_FEAT;
  float*    outp = (float*)d_out;
  const float* stepArr = FP(i_step);

  // ---- init u: (64,25,64,64) from x channels mod 3 ----
  const int N0 = 64 * 64 * 64;
  init_u_kernel<<<25 * N0 / BLK, BLK, 0, stream>>>(FP(xIdx), u);

  int C = 25, H = 64, W = 64, Nn = N0, HW = 64 * 64, hwShift = 12;

  for (int t = 0; t < 16; ++t) {
    if (t == 4 || t == 8 || t == 12) {
      int li = t / 4 - 1;
      int Co = 2 * C;
      int Kp = pad4(C);
      const float* lw  = FP(i_lift0 + 4 * li + 0);
      const float* lg  = FP(i_lift0 + 4 * li + 2);
      const float* lbe = FP(i_lift0 + 4 * li + 3);
      if (Kp > C)   // zero-pad u rows [C, Kp)
        zero_fill_kernel<<<ceil_div((Kp - C) * Nn, BLK), BLK, 0, stream>>>(
            u + (size_t)C * Nn, (Kp - C) * Nn);
      int wtot = pad16(Co) * Kp;
      pad_weights_kernel<<<ceil_div(wtot, BLK), BLK, 0, stream>>>(lw, Wp, Co, C, Kp, wtot);
      gemm16_kernel<0><<<gemm_blocks(Co, Nn), BLK, 0, stream>>>(
          Wp, u, h1, Co, Kp, Nn, nullptr, 0, 0);
      bn_stats_kernel<<<dim3(64, Co), BLK, 0, stream>>>(h1, pS, pQ, Co, Nn);
      bn_finalize_kernel<<<1, BLK, 0, stream>>>(pS, pQ, lg, lbe, bnA, bnC, Co, Nn);
      int Hn = H / 2, Wn = W / 2, HWn = HW / 4, Nnew = Nn / 4;
      pool_bn_relu_kernel<<<ceil_div(Co * Nnew, BLK), BLK, 0, stream>>>(
          h1, u, bnA, bnC, Nn, Nnew, HW, HWn, W, Wn);
      C = Co; H = Hn; W = Wn; HW = HWn; Nn = Nnew; hwShift -= 2;
    }
    int CN = C * Nn;
    float* ux  = ws + OFF_DV + 0 * (size_t)CN;
    float* uy  = ws + OFF_DV + 1 * (size_t)CN;
    float* uxx = ws + OFF_DV + 2 * (size_t)CN;
    float* uyy = ws + OFF_DV + 3 * (size_t)CN;
    float* uxy = ws + OFF_DV + 4 * (size_t)CN;

    deriv4_kernel<<<CN / BLK, BLK, 0, stream>>>(u, ux, uy, uxx, uyy, Nn, H, W, HW);
    derivxy_kernel<<<CN / BLK, BLK, 0, stream>>>(uy, uxy, Nn, H, W, HW);
    zero_scales_kernel<<<1, BLK, 0, stream>>>(smax);
    inv_kernel<<<CN / BLK, BLK, 0, stream>>>(u, ux, uy, uxx, uyy, uxy, inv, smax,
                                             C, Nn, hwShift);
    recip_scales_kernel<<<1, BLK, 0, stream>>>(smax, srec);

    const float* w1p = FP(pb + 8 * t + 0);
    const float* g1  = FP(pb + 8 * t + 2);
    const float* be1 = FP(pb + 8 * t + 3);
    const float* w2p = FP(pb + 8 * t + 4);
    const float* g2  = FP(pb + 8 * t + 6);
    const float* be2 = FP(pb + 8 * t + 7);
    int K1 = 6 * C - 1;
    int K1p = pad4(K1);

    // GEMM1: h1 = w1 * (inv, group-scaled)
    zero_fill_kernel<<<ceil_div((K1p - K1) * Nn, BLK), BLK, 0, stream>>>(
        inv + (size_t)K1 * Nn, (K1p - K1) * Nn);
    int w1tot = pad16(43) * K1p;
    pad_weights_kernel<<<ceil_div(w1tot, BLK), BLK, 0, stream>>>(w1p, Wp, 43, K1, K1p, w1tot);
    gemm16_kernel<1><<<gemm_blocks(43, Nn), BLK, 0, stream>>>(
        Wp, inv, h1, 43, K1p, Nn, srec, C, hwShift);
    bn_stats_kernel<<<dim3(64, 43), BLK, 0, stream>>>(h1, pS, pQ, 43, Nn);
    bn_finalize_kernel<<<1, BLK, 0, stream>>>(pS, pQ, g1, be1, bnA, bnC, 43, Nn);

    // BN1+ReLU applied in place; pad row 43 zeroed for GEMM2 (K 43 -> 44)
    bnrelu_inplace_kernel<<<ceil_div(43 * Nn, BLK), BLK, 0, stream>>>(h1, bnA, bnC, Nn, 43 * Nn);
    zero_fill_kernel<<<ceil_div(Nn, BLK), BLK, 0, stream>>>(h1 + (size_t)43 * Nn, Nn);

    int w2tot = pad16(C) * 44;
    pad_weights_kernel<<<ceil_div(w2tot, BLK), BLK, 0, stream>>>(w2p, Wp, C, 43, 44, w2tot);
    gemm16_kernel<0><<<gemm_blocks(C, Nn), BLK, 0, stream>>>(
        Wp, h1, h2, C, 44, Nn, nullptr, 0, 0);
    bn_stats_kernel<<<dim3(64, C), BLK, 0, stream>>>(h2, pS, pQ, C, Nn);
    bn_finalize_kernel<<<1, BLK, 0, stream>>>(pS, pQ, g2, be2, bnA2, bnC2, C, Nn);

    update_u_kernel<<<CN / BLK, BLK, 0, stream>>>(u, h2, bnA2, bnC2, stepArr, t, Nn);
  }

  // head: global max pool -> fc1 -> relu -> fc2 -> log_softmax
  feat_kernel<<<ceil_div(64 * 200, BLK), BLK, 0, stream>>>(u, feat);
  head_kernel<<<64, 64, 0, stream>>>(feat, FP(i_fc1w), FP(i_fc1b),
                                     FP(i_fc2w), FP(i_fc2b), outp);
}